// RecursiveDecoder_30872224923907
// MI455X (gfx1250) — compile-verified
//
#include <hip/hip_runtime.h>

typedef __attribute__((ext_vector_type(2))) float v2f;
typedef __attribute__((ext_vector_type(8))) float v8f;

#define MF 128
#define HF 256
#define TN 4
#define NSEMF 57
#define WNE_STRIDE (772 * 256)
#define ELD 260   // padded LDS row stride (260 % 64 == 4 -> conflict-free A reads)

// D(16x16) += A(16xK) * B(Kx16), fp32 WMMA, K multiple of 4.
// A: row-major, stride lda. B: row-major (K x N panel), stride ldb.
__device__ __forceinline__ v8f wmma_gemm_tile(const float* A, int lda,
                                              const float* B, int ldb,
                                              int K, int lane) {
  const int ln = lane & 15;
  const int kk = (lane >> 4) << 1;
  v8f c = {0.f, 0.f, 0.f, 0.f, 0.f, 0.f, 0.f, 0.f};
  for (int k0 = 0; k0 < K; k0 += 4) {
    v2f a, b;
    a.x = A[ln * lda + k0 + kk];
    a.y = A[ln * lda + k0 + kk + 1];
    b.x = B[(k0 + kk) * ldb + ln];
    b.y = B[(k0 + kk + 1) * ldb + ln];
    c = __builtin_amdgcn_wmma_f32_16x16x4_f32(false, a, false, b, (short)0, c,
                                              false, false);
  }
  return c;
}

__device__ __forceinline__ void store_tile(float* D, int ldd, v8f c, int lane,
                                           const float* bias, int col0,
                                           bool do_relu) {
  const int col = lane & 15;
  const int r0 = (lane < 16) ? 0 : 8;
  const float bv = bias ? bias[col0 + col] : 0.f;
#pragma unroll
  for (int v = 0; v < 8; ++v) {
    float val = c[v] + bv;
    if (do_relu) val = fmaxf(val, 0.f);
    D[(r0 + v) * ldd + col] = val;
  }
}

// ---------------- K1: child_feats = relu(pf @ W_parent + b) ----------------
__global__ void k_parent(const float* __restrict__ pfp,
                         const float* __restrict__ pfc,
                         const float* __restrict__ pfn,
                         const float* __restrict__ Wp,
                         const float* __restrict__ bp, float* __restrict__ cf0) {
  __shared__ float pf[282];
  for (int r = threadIdx.x; r < 282; r += blockDim.x) {
    float v;
    if (r < 256) v = pfp[r];
    else if (r < 272) v = pfc[r - 256];
    else v = pfn[r - 272];
    pf[r] = v;
  }
  __syncthreads();
  const int k = blockIdx.x * blockDim.x + threadIdx.x;  // 0..32767
  float acc = bp[k];
  for (int r = 0; r < 282; ++r) acc = fmaf(pf[r], Wp[r * 32768 + k], acc);
  cf0[k] = fmaxf(acc, 0.f);
}

// ---------------- K2: P = cf@Wel_a, Q = cf@Wel_b ----------------
__global__ void k_pq(const float* __restrict__ cf0, const float* __restrict__ Wel,
                     float* __restrict__ Pd, float* __restrict__ Qd) {
  const int id = blockIdx.x;  // 0..255
  const int lane = threadIdx.x;
  const int isQ = id >> 7;
  const int t = id & 127;
  const int tm = t >> 4, tn = t & 15;
  v8f c = wmma_gemm_tile(cf0 + tm * 16 * HF, HF,
                         Wel + isQ * HF * HF + tn * 16, HF, HF, lane);
  store_tile((isQ ? Qd : Pd) + tm * 16 * HF + tn * 16, HF, c, lane, nullptr, 0,
             false);
}

// ---------------- exists logits + mask ----------------
__global__ void k_exists(const float* __restrict__ cf0,
                         const float* __restrict__ Wex,
                         const float* __restrict__ bex,
                         float* __restrict__ out_logits,
                         float* __restrict__ existsf) {
  const int m = threadIdx.x;
  if (m < MF) {
    float acc = bex[0];
    for (int r = 0; r < HF; ++r) acc = fmaf(cf0[m * HF + r], Wex[r], acc);
    out_logits[m] = acc;
    existsf[m] = (acc > 0.f) ? 1.f : 0.f;
  }
}

// ---------------- per-iteration: A = cf@Wa, B = cf@Wb, zero cf_next --------
__global__ void k_iterpre(const float* __restrict__ cfprev,
                          const float* __restrict__ Wne,  // W_ne[it] base
                          float* __restrict__ Ad, float* __restrict__ Bd,
                          float* __restrict__ cfnext) {
  const int id = blockIdx.x;  // 0..255
  const int lane = threadIdx.x;
#pragma unroll
  for (int z = 0; z < 4; ++z) cfnext[id * 128 + z * 32 + lane] = 0.f;
  const int isB = id >> 7;
  const int t = id & 127;
  const int tm = t >> 4, tn = t & 15;
  v8f c = wmma_gemm_tile(cfprev + tm * 16 * HF, HF,
                         Wne + isB * HF * HF + tn * 16, HF, HF, lane);
  store_tile((isB ? Bd : Ad) + tm * 16 * HF + tn * 16, HF, c, lane, nullptr, 0,
             false);
}

// ---------------- fused edge kernel: EL tile, logits, EL@Wc, masked msg,
// scatter-max into cf_next ----------------
__global__ void k_edge(const float* __restrict__ Pd, const float* __restrict__ Qd,
                       const float* __restrict__ bel,
                       const float* __restrict__ Wee,
                       const float* __restrict__ bee,
                       const float* __restrict__ Wne,  // W_ne[it] base
                       const float* __restrict__ bne,  // b_ne[it]
                       const float* __restrict__ Ad, const float* __restrict__ Bd,
                       const float* __restrict__ existsf,
                       float* __restrict__ elog_out, int write_logits,
                       unsigned* __restrict__ cfnext_bits) {
  __shared__ float sEL[16 * ELD];
  __shared__ float sPi[HF];
  __shared__ float sWd[TN * HF];
  __shared__ float sLg[16 * TN];

  const int i = blockIdx.y;
  const int j0 = blockIdx.x * 16;
  const int tid = threadIdx.x;  // 128 threads = 4 waves

  for (int s = tid; s < HF; s += 128) sPi[s] = Pd[i * HF + s];
  for (int s = tid; s < TN * HF; s += 128) sWd[s] = Wne[768 * HF + s];
  __syncthreads();

  // EL[jj][h] = relu(P[i][h] + Q[j0+jj][h] + b_el[h])
  for (int e = tid; e < 16 * HF; e += 128) {
    const int jj = e >> 8, h = e & 255;
    float v = sPi[h] + Qd[(j0 + jj) * HF + h] + bel[h];
    sEL[jj * ELD + h] = fmaxf(v, 0.f);
  }
  __syncthreads();

  // edge_exists_logits[i, j0+jj, tt] = EL . Wee[tt] + bee[tt]
  if (tid < 64) {
    const int jj = tid >> 2, tt = tid & 3;
    float acc = bee[tt];
    for (int h = 0; h < HF; ++h)
      acc = fmaf(sEL[jj * ELD + h], Wee[tt * HF + h], acc);
    sLg[jj * TN + tt] = acc;
    if (write_logits) elog_out[(i * MF + j0 + jj) * TN + tt] = acc;
  }
  __syncthreads();

  // D = EL(16x256) @ Wne_c(256x256), fused masked-relu epilogue + scatter max
  const int wave = tid >> 5;
  const int lane = tid & 31;
  const int ln = lane & 15;
  const float exI = existsf[i];
  for (int nt = wave * 4; nt < wave * 4 + 4; ++nt) {
    v8f c = wmma_gemm_tile(sEL, ELD, Wne + 512 * HF + nt * 16, HF, HF, lane);
    const int hcol = nt * 16 + ln;
    const float aI = Ad[i * HF + hcol] + bne[hcol];
    const int r0 = (lane < 16) ? 0 : 8;
    float mx = 0.f;
#pragma unroll
    for (int v = 0; v < 8; ++v) {
      const int jj = r0 + v;
      const int j = j0 + jj;
      const float base = c[v] + aI + Bd[j * HF + hcol];
      const float act = exI * existsf[j];
#pragma unroll
      for (int tt = 0; tt < TN; ++tt) {
        const float lg = sLg[jj * TN + tt];
        if (lg > 0.f && act > 0.f) {
          float cand = fmaxf(base + lg * sWd[tt * HF + hcol], 0.f);
          mx = fmaxf(mx, cand);
        }
      }
    }
    // relu output >= 0 -> uint bit-pattern order matches float order
    atomicMax(&cfnext_bits[i * HF + hcol], __float_as_uint(mx));
  }
}

// ---------------- head: h = relu([cf0|cf1|cf2] @ W_child + b) --------------
__global__ void k_head(const float* __restrict__ cf0, const float* __restrict__ cf1,
                       const float* __restrict__ cf2,
                       const float* __restrict__ Wch,
                       const float* __restrict__ bch, float* __restrict__ hbuf) {
  const int id = blockIdx.x;  // 0..127
  const int lane = threadIdx.x;
  const int tm = id >> 4, tn = id & 15;
  const int ln = lane & 15;
  const int kk = (lane >> 4) << 1;
  v8f c = {0.f, 0.f, 0.f, 0.f, 0.f, 0.f, 0.f, 0.f};
  for (int k0 = 0; k0 < 768; k0 += 4) {
    const float* seg = (k0 < 256) ? cf0 : ((k0 < 512) ? cf1 : cf2);
    const int kl = k0 & 255;
    v2f a, b;
    a.x = seg[(tm * 16 + ln) * HF + kl + kk];
    a.y = seg[(tm * 16 + ln) * HF + kl + kk + 1];
    b.x = Wch[(k0 + kk) * HF + tn * 16 + ln];
    b.y = Wch[(k0 + kk + 1) * HF + tn * 16 + ln];
    c = __builtin_amdgcn_wmma_f32_16x16x4_f32(false, a, false, b, (short)0, c,
                                              false, false);
  }
  store_tile(hbuf + tm * 16 * HF + tn * 16, HF, c, lane, bch, tn * 16, true);
}

// ---------------- child_out = relu(h @ W_child2 + b) ----------------
__global__ void k_out2(const float* __restrict__ hbuf,
                       const float* __restrict__ Wch2,
                       const float* __restrict__ bch2, float* __restrict__ out) {
  const int id = blockIdx.x;
  const int lane = threadIdx.x;
  const int tm = id >> 4, tn = id & 15;
  v8f c = wmma_gemm_tile(hbuf + tm * 16 * HF, HF, Wch2 + tn * 16, HF, HF, lane);
  store_tile(out + tm * 16 * HF + tn * 16, HF, c, lane, bch2, tn * 16, true);
}

// ---------------- child_sem_logits = h @ W_sem + b (57 cols, scalar) -------
__global__ void k_sem(const float* __restrict__ hbuf,
                      const float* __restrict__ Wsem,
                      const float* __restrict__ bsem,
                      float* __restrict__ out_sem) {
  const int id = blockIdx.x * blockDim.x + threadIdx.x;
  if (id >= MF * NSEMF) return;
  const int m = id / NSEMF, s = id - m * NSEMF;
  float acc = bsem[s];
  for (int r = 0; r < HF; ++r)
    acc = fmaf(hbuf[m * HF + r], Wsem[r * NSEMF + s], acc);
  out_sem[m * NSEMF + s] = acc;
}

extern "C" void kernel_launch(void* const* d_in, const int* in_sizes, int n_in,
                              void* d_out, int out_size, void* d_ws,
                              size_t ws_size, hipStream_t stream) {
  const float* parent = (const float*)d_in[0];
  const float* gchild = (const float*)d_in[1];
  const float* gnum   = (const float*)d_in[2];
  const float* Wp     = (const float*)d_in[3];
  const float* bp     = (const float*)d_in[4];
  const float* Wex    = (const float*)d_in[5];
  const float* bex    = (const float*)d_in[6];
  const float* Wsem   = (const float*)d_in[7];
  const float* bsem   = (const float*)d_in[8];
  const float* Wel    = (const float*)d_in[9];
  const float* bel    = (const float*)d_in[10];
  const float* Wee    = (const float*)d_in[11];
  const float* bee    = (const float*)d_in[12];
  const float* Wne    = (const float*)d_in[13];
  const float* bne    = (const float*)d_in[14];
  const float* Wch    = (const float*)d_in[15];
  const float* bch    = (const float*)d_in[16];
  const float* Wch2   = (const float*)d_in[17];
  const float* bch2   = (const float*)d_in[18];

  float* out = (float*)d_out;
  float* o_child = out;          // [1,128,256] = 32768
  float* o_sem   = out + 32768;  // [1,128,57]  = 7296
  float* o_ex    = out + 40064;  // [1,128,1]   = 128
  float* o_elog  = out + 40192;  // [1,128,128,4] = 65536

  float* ws  = (float*)d_ws;
  float* cf0 = ws;            // 32768
  float* Pd  = ws + 32768;    // 32768
  float* Qd  = ws + 65536;    // 32768
  float* exf = ws + 98304;    // 128
  float* cf1 = ws + 98432;    // 32768
  float* cf2 = ws + 131200;   // 32768
  float* Ad  = ws + 163968;   // 32768
  float* Bd  = ws + 196736;   // 32768
  float* hb  = ws + 229504;   // 32768  (end: 262272 floats ~= 1.0 MB)

  k_parent<<<128, 256, 0, stream>>>(parent, gchild, gnum, Wp, bp, cf0);
  k_pq<<<256, 32, 0, stream>>>(cf0, Wel, Pd, Qd);
  k_exists<<<1, 128, 0, stream>>>(cf0, Wex, bex, o_ex, exf);

  // iteration 0 (cf0 -> cf1)
  k_iterpre<<<256, 32, 0, stream>>>(cf0, Wne + 0 * WNE_STRIDE, Ad, Bd, cf1);
  k_edge<<<dim3(8, 128), 128, 0, stream>>>(Pd, Qd, bel, Wee, bee,
                                           Wne + 0 * WNE_STRIDE, bne + 0 * 256,
                                           Ad, Bd, exf, o_elog, 1,
                                           (unsigned*)cf1);
  // iteration 1 (cf1 -> cf2)
  k_iterpre<<<256, 32, 0, stream>>>(cf1, Wne + 1 * WNE_STRIDE, Ad, Bd, cf2);
  k_edge<<<dim3(8, 128), 128, 0, stream>>>(Pd, Qd, bel, Wee, bee,
                                           Wne + 1 * WNE_STRIDE, bne + 1 * 256,
                                           Ad, Bd, exf, o_elog, 0,
                                           (unsigned*)cf2);

  k_head<<<128, 32, 0, stream>>>(cf0, cf1, cf2, Wch, bch, hb);
  k_out2<<<128, 32, 0, stream>>>(hb, Wch2, bch2, o_child);
  k_sem<<<29, 256, 0, stream>>>(hb, Wsem, bsem, o_sem);
}